// DecoderSyntaxTreeGRU_82136954569228
// MI455X (gfx1250) — compile-verified
//
#include <hip/hip_runtime.h>
#include <math.h>

// Problem constants (match reference)
#define BATCH 8
#define BP    16      // batch padded to WMMA N=16
#define HDIM  256
#define VOCAB 32000
#define DEPTH 8
#define NLEAF 256

typedef __attribute__((ext_vector_type(2))) float v2f;
typedef __attribute__((ext_vector_type(8))) float v8f;
typedef __attribute__((ext_vector_type(4))) unsigned int u32x4;
typedef __attribute__((ext_vector_type(8))) int i32x8;
typedef __attribute__((ext_vector_type(4))) int i32x4;

// D = A(16x4, f32) * B(4x16, f32) + C(16x16, f32)   -- native fp32 WMMA
__device__ __forceinline__ v8f wmma_f32_k4(v2f a, v2f b, v8f c) {
    return __builtin_amdgcn_wmma_f32_16x16x4_f32(
        /*neg_a=*/false, a, /*neg_b=*/false, b,
        /*c_mod=*/(short)0, c, /*reuse_a=*/false, /*reuse_b=*/false);
}

// ---------------------------------------------------------------------------
// Pair-interleaved h layout (per node, 4096 floats = 16 KB):
//   element (k, b) lives at  (k>>1)*32 + b*2 + (k&1)
// A lane's WMMA B-operand pair (k, k+1) at batch column b is therefore a
// single 8-byte contiguous v2f, and a 16-lane half-wave reads one contiguous
// 128-byte segment -> single global_load_b64 / ds_load_b64 per lane.
// ---------------------------------------------------------------------------
#define PIDX(k, b) ((((k) >> 1) * BP + (b)) * 2 + ((k) & 1))

__global__ void init_h_kernel(const float* __restrict__ enc, float* __restrict__ hT2) {
    const int k = threadIdx.x;          // 256 threads, one per k-row
    #pragma unroll
    for (int b = 0; b < BP; ++b)
        hT2[PIDX(k, b)] = (b < BATCH) ? enc[b * HDIM + k] : 0.0f;
}

// ---------------------------------------------------------------------------
// TDM: DMA one 16 KB pair-interleaved h tile from global into LDS.
// D# per CDNA5 ISA 8.3-8.4: group0 = {count=1, lds_addr, global_addr, type=2},
// group1 = {data_size=4B, tensor_dim0=4096, tensor_dim1=1, tile_dim0=4096,
//           tile_dim1=1, tensor_dim0_stride=4096}. Groups 2/3 unused (zero).
// This toolchain uses the 6-arg builtin form (extra int32x8 group + cpol).
// ---------------------------------------------------------------------------
__device__ __forceinline__ void tdm_copy_htile_to_lds(const float* gsrc, unsigned lds_off) {
    const unsigned long long ga = (unsigned long long)(uintptr_t)gsrc;
    u32x4 g0;
    g0[0] = 1u;                                              // count = 1 valid descriptor
    g0[1] = lds_off;                                         // LDS byte address
    g0[2] = (unsigned)(ga & 0xFFFFFFFFu);                    // global_addr[31:0]
    g0[3] = (unsigned)((ga >> 32) & 0x01FFFFFFu) | (2u << 30); // addr[56:32] | type=2
    i32x8 g1;
    g1[0] = (int)(2u << 16);        // data_size = 2 (4 bytes), workgroup_mask = 0
    g1[1] = (int)(4096u << 16);     // tensor_dim0[15:0] << 16 (barrier addr = 0)
    g1[2] = (int)(1u << 16);        // tensor_dim0[31:16]=0 | tensor_dim1 = 1
    g1[3] = (int)(4096u << 16);     // tensor_dim1[31:16]=0 | tile_dim0 = 4096
    g1[4] = 1;                      // tile_dim1 = 1, tile_dim2 = 0
    g1[5] = 4096;                   // tensor_dim0_stride[31:0]
    g1[6] = 0;                      // stride[47:32] = 0, tensor_dim1_stride[15:0] = 0
    g1[7] = 0;                      // tensor_dim1_stride[47:16] = 0
    i32x4 gz4 = {0, 0, 0, 0};
    i32x8 gz8 = {0, 0, 0, 0, 0, 0, 0, 0};
    __builtin_amdgcn_tensor_load_to_lds(g0, g1, gz4, gz4, gz8, 0);
}

// ---------------------------------------------------------------------------
// Software-pipelined dual-tile GEMM core: one wave computes two 16x16 output
// tiles (acc0/acc1, M rows rt0/rt1) sharing the B operand; next-step A0/A1/B
// are loaded into fresh registers before the current WMMAs consume theirs.
// ---------------------------------------------------------------------------
#define GEMM_DUAL(wrow0, wrow1, bp, kb, acc0, acc1)                              \
    {                                                                            \
        v2f a0 = *(const v2f*)((wrow0) + (kb));                                  \
        v2f a1 = *(const v2f*)((wrow1) + (kb));                                  \
        v2f bb = *(const v2f*)((bp) + (((kb) >> 1) * (BP * 2)));                 \
        _Pragma("unroll 4")                                                      \
        for (int k0 = 0; k0 < HDIM - 4; k0 += 4) {                               \
            v2f a0n = *(const v2f*)((wrow0) + k0 + 4 + (kb));                    \
            v2f a1n = *(const v2f*)((wrow1) + k0 + 4 + (kb));                    \
            v2f bbn = *(const v2f*)((bp) + (((k0 + 4 + (kb)) >> 1) * (BP * 2))); \
            acc0 = wmma_f32_k4(a0, bb, acc0);                                    \
            acc1 = wmma_f32_k4(a1, bb, acc1);                                    \
            a0 = a0n; a1 = a1n; bb = bbn;                                        \
        }                                                                        \
        acc0 = wmma_f32_k4(a0, bb, acc0);                                        \
        acc1 = wmma_f32_k4(a1, bb, acc1);                                        \
    }

// ---------------------------------------------------------------------------
// Stage A: r,z gates.  blockIdx.x = node*2+child, blockIdx.y = gate (0=r,1=z)
// 256 threads = 8 waves; wave w owns M-tiles {w, w+8}.
//   gate 0: rh[nc][k][b] = sigmoid(acc) * hT_parent[node][k][b]
//   gate 1: zb[nc][k][b] = sigmoid(acc)
// ---------------------------------------------------------------------------
__global__ void gru_rz_kernel(const float* __restrict__ gru_w,
                              const int*   __restrict__ rules,
                              const float* __restrict__ hTp,
                              float* __restrict__ rh,
                              float* __restrict__ zb,
                              int rule_off) {
    const int nc   = blockIdx.x;
    const int gate = blockIdx.y;
    const int node = nc >> 1, c = nc & 1;
    const int lane = threadIdx.x & 31;
    const int wave = threadIdx.x >> 5;
    const int rule = rules[rule_off + node];

    const float* Wg = gru_w + ((size_t)((rule * 2 + c) * 3 + gate)) * HDIM * HDIM;
    const float* hp = hTp + (size_t)node * HDIM * BP;
    float* dst = (gate == 0 ? rh : zb) + (size_t)nc * HDIM * BP;

    const int rlo = lane & 15;
    const int kb  = (lane >> 4) * 2;   // fp32 A/B operand: hi half-lanes hold K+2,K+3
    const int mhi = (lane >> 4) * 8;   // C/D: hi half-lanes hold M+8
    const int rt0 = wave, rt1 = wave + 8;

    const float* wrow0 = Wg + (size_t)(rt0 * 16 + rlo) * HDIM;
    const float* wrow1 = Wg + (size_t)(rt1 * 16 + rlo) * HDIM;
    const float* bptr  = hp + rlo * 2;

    v8f acc0 = {}, acc1 = {};
    GEMM_DUAL(wrow0, wrow1, bptr, kb, acc0, acc1);

    #pragma unroll
    for (int i = 0; i < 8; ++i) {
        const int ro0 = rt0 * 16 + i + mhi;
        const int ro1 = rt1 * 16 + i + mhi;
        const int i0 = PIDX(ro0, rlo), i1 = PIDX(ro1, rlo);
        const float s0 = 1.0f / (1.0f + expf(-acc0[i]));
        const float s1 = 1.0f / (1.0f + expf(-acc1[i]));
        if (gate == 0) { dst[i0] = s0 * hp[i0]; dst[i1] = s1 * hp[i1]; }
        else           { dst[i0] = s0;          dst[i1] = s1;          }
    }
}

// ---------------------------------------------------------------------------
// Stage B: ng = tanh(W2 @ (r*h)); h_child = (1-z)*ng + z*h_parent + bias
// blockIdx.x = node*2+child
// ---------------------------------------------------------------------------
__global__ void gru_ng_kernel(const float* __restrict__ gru_w,
                              const int*   __restrict__ rules,
                              const int*   __restrict__ cbidx,
                              const float* __restrict__ nt_bias,
                              const float* __restrict__ hTp,
                              const float* __restrict__ rh,
                              const float* __restrict__ zb,
                              float* __restrict__ hTc,
                              int rule_off, int bias_off) {
    const int nc   = blockIdx.x;
    const int node = nc >> 1, c = nc & 1;
    const int lane = threadIdx.x & 31;
    const int wave = threadIdx.x >> 5;
    const int rule = rules[rule_off + node];
    const int bidx = cbidx[bias_off + nc];

    const float* W2  = gru_w + ((size_t)((rule * 2 + c) * 3 + 2)) * HDIM * HDIM;
    const float* hp  = hTp + (size_t)node * HDIM * BP;
    const float* rhn = rh  + (size_t)nc   * HDIM * BP;
    const float* zn  = zb  + (size_t)nc   * HDIM * BP;
    float*       out = hTc + (size_t)nc   * HDIM * BP;
    const float* bv  = nt_bias + (size_t)bidx * HDIM;

    const int rlo = lane & 15;
    const int kb  = (lane >> 4) * 2;
    const int mhi = (lane >> 4) * 8;
    const int rt0 = wave, rt1 = wave + 8;

    const float* wrow0 = W2 + (size_t)(rt0 * 16 + rlo) * HDIM;
    const float* wrow1 = W2 + (size_t)(rt1 * 16 + rlo) * HDIM;
    const float* bptr  = rhn + rlo * 2;

    v8f acc0 = {}, acc1 = {};
    GEMM_DUAL(wrow0, wrow1, bptr, kb, acc0, acc1);

    #pragma unroll
    for (int i = 0; i < 8; ++i) {
        const int ro0 = rt0 * 16 + i + mhi;
        const int ro1 = rt1 * 16 + i + mhi;
        const int i0 = PIDX(ro0, rlo), i1 = PIDX(ro1, rlo);
        const float z0 = zn[i0], z1 = zn[i1];
        out[i0] = (1.0f - z0) * tanhf(acc0[i]) + z0 * hp[i0] + bv[ro0];
        out[i1] = (1.0f - z1) * tanhf(acc1[i]) + z1 * hp[i1] + bv[ro1];
    }
}

// ---------------------------------------------------------------------------
// Leaf projection: logits[l,b,v] = sum_k term_w[rule_l, v, k] * h[l,b,k]
// grid = (NLEAF, VOCAB/256); leaf index fastest so all 256 leaves reuse the
// same term_w v-rows while they are L2-resident. The leaf's 16 KB h tile is
// DMA'd into LDS once per block by the Tensor Data Mover; all 8 waves then
// feed their WMMA B-operands from LDS (ds_load_b64, conflict-free).
// Wave w owns v-tiles {w, w+8} (256 v per block).
// ---------------------------------------------------------------------------
__global__ void leaf_gemm_kernel(const float* __restrict__ term_w,
                                 const int*   __restrict__ leaf_rules,
                                 const float* __restrict__ hTl,
                                 float* __restrict__ out) {
    __shared__ float hs[HDIM * BP];   // 16 KB pair-interleaved h tile

    const int leaf   = blockIdx.x;
    const int vgroup = blockIdx.y;
    const int lane   = threadIdx.x & 31;
    const int wave   = threadIdx.x >> 5;
    const int rule   = leaf_rules[leaf];

    const float* hp = hTl + (size_t)leaf * HDIM * BP;
    if (wave == 0) {
        tdm_copy_htile_to_lds(hp, (unsigned)(uintptr_t)(void*)hs);
        __builtin_amdgcn_s_wait_tensorcnt(0);
    }
    __syncthreads();

    const int rlo = lane & 15;
    const int kb  = (lane >> 4) * 2;
    const int mhi = (lane >> 4) * 8;

    const int vbase0 = vgroup * 256 + wave * 16;
    const int vbase1 = vbase0 + 128;          // (wave+8)*16

    const float* wrow0 = term_w + ((size_t)rule * VOCAB + vbase0 + rlo) * HDIM;
    const float* wrow1 = term_w + ((size_t)rule * VOCAB + vbase1 + rlo) * HDIM;
    const float* bptr  = hs + rlo * 2;

    v8f acc0 = {}, acc1 = {};
    GEMM_DUAL(wrow0, wrow1, bptr, kb, acc0, acc1);

    const int bcol = rlo;                 // N index = batch
    if (bcol < BATCH) {
        float* orow = out + ((size_t)leaf * BATCH + bcol) * VOCAB;
        #pragma unroll
        for (int i = 0; i < 8; ++i) {
            orow[vbase0 + i + mhi] = acc0[i];
            orow[vbase1 + i + mhi] = acc1[i];
        }
    }
}

// ---------------------------------------------------------------------------
// In-place softmax over v for each (leaf, batch) row of 32000.
// ---------------------------------------------------------------------------
__global__ void softmax_rows_kernel(float* __restrict__ out) {
    float* p = out + (size_t)blockIdx.x * VOCAB;
    const int tid = threadIdx.x;
    __shared__ float red[256];

    float m = -INFINITY;
    for (int v = tid; v < VOCAB; v += 256) m = fmaxf(m, p[v]);
    red[tid] = m; __syncthreads();
    for (int s = 128; s > 0; s >>= 1) {
        if (tid < s) red[tid] = fmaxf(red[tid], red[tid + s]);
        __syncthreads();
    }
    m = red[0]; __syncthreads();

    float sum = 0.0f;
    for (int v = tid; v < VOCAB; v += 256) sum += expf(p[v] - m);
    red[tid] = sum; __syncthreads();
    for (int s = 128; s > 0; s >>= 1) {
        if (tid < s) red[tid] += red[tid + s];
        __syncthreads();
    }
    const float inv = 1.0f / red[0];

    for (int v = tid; v < VOCAB; v += 256) p[v] = expf(p[v] - m) * inv;
}

// ---------------------------------------------------------------------------
extern "C" void kernel_launch(void* const* d_in, const int* in_sizes, int n_in,
                              void* d_out, int out_size, void* d_ws, size_t ws_size,
                              hipStream_t stream) {
    const float* enc            = (const float*)d_in[0];
    const float* gru_w          = (const float*)d_in[1];
    const float* term_w         = (const float*)d_in[2];
    const float* nt_bias        = (const float*)d_in[3];
    const int*   internal_rules = (const int*)  d_in[4];
    const int*   child_bias_idx = (const int*)  d_in[5];
    const int*   leaf_rules     = (const int*)  d_in[6];
    float* out = (float*)d_out;

    // workspace: 4 buffers of 256 nodes * H * BP floats (4 MB each, 16 MB total)
    const size_t BUF = (size_t)NLEAF * HDIM * BP;   // 1<<20 floats
    float* ws = (float*)d_ws;
    float* hA = ws;
    float* hB = ws + BUF;
    float* rh = ws + 2 * BUF;
    float* zb = ws + 3 * BUF;

    init_h_kernel<<<1, 256, 0, stream>>>(enc, hA);

    float* parent = hA;
    float* child  = hB;
    for (int d = 0; d < DEPTH; ++d) {
        const int n = 1 << d;
        const int rule_off = n - 1;       // sum_{k<d} 2^k
        const int bias_off = 2 * n - 2;   // sum_{k<d} 2^(k+1)
        dim3 gA(n * 2, 2);
        gru_rz_kernel<<<gA, 256, 0, stream>>>(gru_w, internal_rules, parent,
                                              rh, zb, rule_off);
        gru_ng_kernel<<<n * 2, 256, 0, stream>>>(gru_w, internal_rules,
                                                 child_bias_idx, nt_bias,
                                                 parent, rh, zb, child,
                                                 rule_off, bias_off);
        float* t = parent; parent = child; child = t;
    }
    // After 8 levels, leaves live in `parent` (== hA).

    dim3 gL(NLEAF, VOCAB / 256);   // (256, 125)
    leaf_gemm_kernel<<<gL, 256, 0, stream>>>(term_w, leaf_rules, parent, out);

    softmax_rows_kernel<<<NLEAF * BATCH, 256, 0, stream>>>(out);
}